// TinyRNN_72670846648717
// MI455X (gfx1250) — compile-verified
//
#include <hip/hip_runtime.h>

// ---------------------------------------------------------------------------
// TinyGRU on MI455X (gfx1250), WMMA-based recurrence.
//
// Shapes: I=3, H=4, O=2, B=4096, T=2048.  Recurrence serializes T; the 4096
// sequences are the only parallelism.  We process 16 sequences per wave with
// V_WMMA_F32_16X16X4_F32, computing the transposed products so the A operands
// (W_in^T, W_h^T padded to 16x4) are loop-invariant:
//
//   fin^T = W_in^T (16x4) x x^T (4x16) + b_in   (one WMMA)
//   fh^T  = W_h^T  (16x4) x h^T (4x16) + b_h    (one WMMA)
//
// D layout (16x16 f32): lane n holds rows 0..7 (r,z pre-acts) for seq n;
// lane n+16 holds rows 8..15 (n-gate pre-acts).  Gate math needs only fixed
// half-wave exchanges (shfl_xor 16).  Both lane halves maintain a full copy
// of h[0..3], so the next step's B operand is 2 v_cndmask, and each half
// stores its own decoder output column (branch-free, full-EXEC).
// ---------------------------------------------------------------------------

typedef float v2f __attribute__((ext_vector_type(2)));
typedef float v8f __attribute__((ext_vector_type(8)));

#if defined(__HIP_DEVICE_COMPILE__) && !__has_builtin(__builtin_amdgcn_wmma_f32_16x16x4_f32)
#error "gfx1250 WMMA f32 16x16x4 builtin not available"
#endif

namespace {
constexpr int Bn = 4096;
constexpr int Tn = 2048;
constexpr int G3 = 12;   // 3*H
constexpr int On = 2;
}

__device__ __forceinline__ float fsig(float x) {
  // sigmoid(x) = 1 / (1 + 2^(-x*log2e))   -> v_exp_f32 + v_rcp_f32 (TRANS32)
  float e = __builtin_amdgcn_exp2f(x * -1.4426950408889634f);
  return __builtin_amdgcn_rcpf(1.0f + e);
}
__device__ __forceinline__ float ftanh(float x) {
  // tanh(x) = (1 - 2^(-2x*log2e)) / (1 + 2^(-2x*log2e))
  float e = __builtin_amdgcn_exp2f(x * -2.8853901817779268f);
  return (1.0f - e) * __builtin_amdgcn_rcpf(1.0f + e);
}

__global__ __launch_bounds__(32) void gru_wmma_kernel(
    const float* __restrict__ in,    // (B, T, 3)
    const float* __restrict__ Win,   // (3, 12)
    const float* __restrict__ Wh,    // (4, 12)
    const float* __restrict__ bias,  // (24,)
    const float* __restrict__ dW,    // (4, 2)
    const float* __restrict__ db,    // (2,)
    float* __restrict__ out)         // (B, T, 2)
{
  const int lane = threadIdx.x & 31;
  const int m    = lane & 15;          // row / sequence-in-block
  const bool lo  = lane < 16;          // lane half
  const int seq  = blockIdx.x * 16 + m;
  const bool gv  = m < G3;             // valid gate column (12 of 16)

  // ---- A operands: W_in^T and W_h^T in 16x4 A layout (loop-invariant) ----
  // lane m:    v0 = A[m][0], v1 = A[m][1];  lane m+16: v0 = A[m][2], v1 = A[m][3]
  v2f aWin, aWh;
  aWin.x = gv ? Win[(lo ? 0 : 2) * G3 + m] : 0.0f;
  aWin.y = (gv && lo) ? Win[1 * G3 + m] : 0.0f;     // K=3 row of W_in^T is pad
  aWh.x  = gv ? Wh[(lo ? 0 : 2) * G3 + m] : 0.0f;
  aWh.y  = gv ? Wh[(lo ? 1 : 3) * G3 + m] : 0.0f;

  // ---- bias C operands in 16x16 f32 C/D layout: C[row][*] = bias[row] ----
  v8f cbin, cbh;
#pragma unroll
  for (int r = 0; r < 8; ++r) {
    int row = (lo ? 0 : 8) + r;
    cbin[r] = (row < G3) ? bias[row] : 0.0f;        // b_in
    cbh[r]  = (row < G3) ? bias[G3 + row] : 0.0f;   // b_h
  }

  // ---- decoder: each lane half owns one output column ----
  const int oc = lo ? 0 : 1;
  const float dw0 = dW[0 * On + oc], dw1 = dW[1 * On + oc];
  const float dw2 = dW[2 * On + oc], dw3 = dW[3 * On + oc];
  const float dcb = db[oc];

  // ---- per-lane streaming pointers ----
  // lane n reads x0 (pa) and x1 (pb); lane n+16 reads x2 (pa) and re-reads x2
  // (pb aliases pa -> L0 hit, no extra DRAM traffic), discarded via cndmask.
  const float* pa = in + (size_t)seq * (3 * Tn) + (lo ? 0 : 2);
  const float* pb = in + (size_t)seq * (3 * Tn) + (lo ? 1 : 2);
  float* po = out + (size_t)seq * (On * Tn) + oc;

  float h0 = 0.0f, h1 = 0.0f, h2 = 0.0f, h3 = 0.0f;

#pragma unroll 4
  for (int t = 0; t < Tn; ++t) {
    const float xa = pa[3 * t];
    const float xb = pb[3 * t];

    // B operands in 4x16 B layout: lane n: {B[0][n],B[1][n]}, lane n+16: {B[2][n],B[3][n]}
    v2f bx; bx.x = xa;            bx.y = lo ? xb : 0.0f;   // x^T (K=3 row pad)
    v2f bh; bh.x = lo ? h0 : h2;  bh.y = lo ? h1 : h3;     // h^T

    // fin^T = W_in^T * x^T + b_in ;  fh^T = W_h^T * h^T + b_h
    v8f dfin = __builtin_amdgcn_wmma_f32_16x16x4_f32(
        false, aWin, false, bx, (short)0, cbin, false, false);
    v8f dfh = __builtin_amdgcn_wmma_f32_16x16x4_f32(
        false, aWh, false, bh, (short)0, cbh, false, false);

    // lanes 0-15: rows 0..3 = r pre-acts, 4..7 = z pre-acts (valid here)
    // lanes 16-31: rows 8..11 = n-gate pre-acts (valid there)
    float rz[8];
#pragma unroll
    for (int i = 0; i < 8; ++i) rz[i] = fsig(dfin[i] + dfh[i]);

    // half-wave exchanges: send r,z down to lanes 16-31
    float rX[4], zX[4];
#pragma unroll
    for (int j = 0; j < 4; ++j) rX[j] = __shfl_xor(rz[j], 16, 32);
#pragma unroll
    for (int j = 0; j < 4; ++j) zX[j] = __shfl_xor(rz[4 + j], 16, 32);

    // n-gate (valid in lanes 16-31): n = tanh(fin_n + r * fh_n)
    float nn[4], nX[4];
#pragma unroll
    for (int j = 0; j < 4; ++j) nn[j] = ftanh(dfin[j] + rX[j] * dfh[j]);
#pragma unroll
    for (int j = 0; j < 4; ++j) nX[j] = __shfl_xor(nn[j], 16, 32);

    // both halves compute identical h_new = n + z*(h - n)
    const float z0 = lo ? rz[4] : zX[0], n0 = lo ? nX[0] : nn[0];
    const float z1 = lo ? rz[5] : zX[1], n1 = lo ? nX[1] : nn[1];
    const float z2 = lo ? rz[6] : zX[2], n2 = lo ? nX[2] : nn[2];
    const float z3 = lo ? rz[7] : zX[3], n3 = lo ? nX[3] : nn[3];
    h0 = n0 + z0 * (h0 - n0);
    h1 = n1 + z1 * (h1 - n1);
    h2 = n2 + z2 * (h2 - n2);
    h3 = n3 + z3 * (h3 - n3);

    // decoder: each half stores its own output column (full-EXEC b32 store)
    po[On * t] = dcb + h0 * dw0 + h1 * dw1 + h2 * dw2 + h3 * dw3;
  }
}

extern "C" void kernel_launch(void* const* d_in, const int* in_sizes, int n_in,
                              void* d_out, int out_size, void* d_ws, size_t ws_size,
                              hipStream_t stream) {
  (void)in_sizes; (void)n_in; (void)out_size; (void)d_ws; (void)ws_size;
  const float* inputs = (const float*)d_in[0];
  const float* W_in   = (const float*)d_in[1];
  const float* W_h    = (const float*)d_in[2];
  const float* bias   = (const float*)d_in[3];
  const float* dec_W  = (const float*)d_in[4];
  const float* dec_b  = (const float*)d_in[5];
  float* out = (float*)d_out;

  dim3 grid(Bn / 16);   // 256 single-wave workgroups -> max spread over WGPs
  dim3 block(32);       // one wave32 per block, 16 sequences per wave
  gru_wmma_kernel<<<grid, block, 0, stream>>>(inputs, W_in, W_h, bias,
                                              dec_W, dec_b, out);
}